// LinearMultiheadSplit_64802466562905
// MI455X (gfx1250) — compile-verified
//
#include <hip/hip_runtime.h>

#define IN_F     512
#define OUT_F    512
#define N_HEADS  32
#define N_SPLITS 4
#define N_GROUPS (N_HEADS * N_SPLITS)   // 128
#define NSAMP    1024
#define DELTA_SCALE 0.1f
#define MAX_TILES 192                   // sum ceil(c_g/16) <= 64 + 128 (safe bound)
#define XS_STRIDE 520                   // 512 + 8 pad floats: breaks LDS bank aliasing

// ---- workspace layout (ints) ----
// [0,128)     counts per group
// [128,256)   scatter cursor per group
// [256,384)   group offsets (exclusive prefix)
// [384]       numTiles
// [400,592)   tileGroup
// [592,784)   tileBase (position in idxList)
// [784,976)   tileLen  (1..16)
// [1024,2048) idxList (gathered sample ids, group-major)

typedef __attribute__((ext_vector_type(2))) float v2f;
typedef __attribute__((ext_vector_type(8))) float v8f;

__global__ void lms_init(int* ws) {
  int i = threadIdx.x + blockIdx.x * blockDim.x;
  if (i < 1024) ws[i] = 0;
}

__global__ void lms_count(const int* __restrict__ head, const int* __restrict__ split,
                          int* ws) {
  int i = threadIdx.x + blockIdx.x * blockDim.x;
  if (i < NSAMP) {
    int g = head[i] * N_SPLITS + split[i];
    atomicAdd(&ws[g], 1);
  }
}

__global__ void lms_plan(int* ws) {
  if (threadIdx.x != 0 || blockIdx.x != 0) return;
  int off = 0, nt = 0;
  for (int g = 0; g < N_GROUPS; ++g) {
    int c = ws[g];
    ws[256 + g] = off;
    ws[128 + g] = off;   // scatter cursor starts at group offset
    for (int m0 = 0; m0 < c; m0 += 16) {
      ws[400 + nt] = g;
      ws[592 + nt] = off + m0;
      ws[784 + nt] = (c - m0 < 16) ? (c - m0) : 16;
      ++nt;
    }
    off += c;
  }
  ws[384] = nt;
}

__global__ void lms_scatter(const int* __restrict__ head, const int* __restrict__ split,
                            int* ws) {
  int i = threadIdx.x + blockIdx.x * blockDim.x;
  if (i < NSAMP) {
    int g = head[i] * N_SPLITS + split[i];
    int pos = atomicAdd(&ws[128 + g], 1);
    ws[1024 + pos] = i;
  }
}

// grid: (MAX_TILES, OUT_F/128); block: 256 threads = 8 wave32, each wave owns 16 cols.
__global__ __launch_bounds__(256) void lms_gemm(
    const float* __restrict__ input,   // [N, IN_F]
    const float* __restrict__ weight,  // [N_HEADS, IN_F, OUT_F]
    const float* __restrict__ delta,   // [N_GROUPS, IN_F, OUT_F]
    const float* __restrict__ bias,    // [N_HEADS, OUT_F]
    const int*   __restrict__ ws,
    float*       __restrict__ out)     // [N, OUT_F]
{
  __shared__ float Xs[16 * XS_STRIDE]; // 16 gathered input rows (padded rows = 0)
  __shared__ int   rowIdx[16];

  const int t = blockIdx.x;
  if (t >= ws[384]) return;

  const int g    = ws[400 + t];
  const int base = ws[592 + t];
  const int len  = ws[784 + t];
  const int h    = g >> 2;
  const int tid  = threadIdx.x;

  if (tid < 16) rowIdx[tid] = (tid < len) ? ws[1024 + base + tid] : -1;
  __syncthreads();

  // Cooperative gather of 16 rows x 512 f32 into LDS (float4 per thread x 8).
  #pragma unroll
  for (int it = 0; it < 8; ++it) {
    int e  = tid + it * 256;        // float4 slot in [0, 2048)
    int r  = e >> 7;                // 128 float4 per row
    int k4 = e & 127;
    int ri = rowIdx[r];
    float4 v = make_float4(0.f, 0.f, 0.f, 0.f);
    if (ri >= 0)
      v = ((const float4*)(input + (size_t)ri * IN_F))[k4];
    *(float4*)(&Xs[r * XS_STRIDE + k4 * 4]) = v;
  }
  __syncthreads();

  const int lane = tid & 31;
  const int wave = tid >> 5;
  const int nIdx = (blockIdx.y * 8 + wave) * 16 + (lane & 15); // output column
  const int koff = (lane >> 4) * 2;   // lanes 0-15 -> K 0,1 ; lanes 16-31 -> K 2,3
  const int m    = lane & 15;         // A-matrix row for this lane

  // Accumulator = broadcast bias (column-dependent only; padded rows never stored).
  const float bv = bias[h * OUT_F + nIdx];
  v8f acc;
  #pragma unroll
  for (int j = 0; j < 8; ++j) acc[j] = bv;

  const float* __restrict__ Wp = weight + (size_t)h * (IN_F * OUT_F) + nIdx;
  const float* __restrict__ Dp = delta  + (size_t)g * (IN_F * OUT_F) + nIdx;
  const float* __restrict__ Xp = &Xs[m * XS_STRIDE];

  for (int kk = 0; kk < IN_F; kk += 4) {
    const int k0 = kk + koff;
    v2f a, b;
    a[0] = Xp[k0];
    a[1] = Xp[k0 + 1];
    // fused effective weight: W[h] + 0.1 * D[g], fp32 end-to-end
    b[0] = Wp[(size_t)k0 * OUT_F]       + DELTA_SCALE * Dp[(size_t)k0 * OUT_F];
    b[1] = Wp[(size_t)(k0 + 1) * OUT_F] + DELTA_SCALE * Dp[(size_t)(k0 + 1) * OUT_F];
    acc = __builtin_amdgcn_wmma_f32_16x16x4_f32(
        /*neg_a=*/false, a, /*neg_b=*/false, b,
        /*c_mod=*/(short)0, acc, /*reuse_a=*/false, /*reuse_b=*/false);
  }

  // C/D layout: VGPR r -> M=r (lanes 0-15) / M=r+8 (lanes 16-31), N = lane&15.
  #pragma unroll
  for (int r = 0; r < 8; ++r) {
    int M  = (lane < 16) ? r : (r + 8);
    int ri = rowIdx[M];
    if (ri >= 0) out[(size_t)ri * OUT_F + nIdx] = acc[r];
  }
}

extern "C" void kernel_launch(void* const* d_in, const int* in_sizes, int n_in,
                              void* d_out, int out_size, void* d_ws, size_t ws_size,
                              hipStream_t stream) {
  const float* input  = (const float*)d_in[0];
  const int*   head   = (const int*)d_in[1];
  const int*   split  = (const int*)d_in[2];
  const float* weight = (const float*)d_in[3];
  const float* delta  = (const float*)d_in[4];
  const float* bias   = (const float*)d_in[5];
  float* out = (float*)d_out;
  int*   ws  = (int*)d_ws;

  lms_init   <<<4, 256, 0, stream>>>(ws);
  lms_count  <<<4, 256, 0, stream>>>(head, split, ws);
  lms_plan   <<<1, 1,   0, stream>>>(ws);
  lms_scatter<<<4, 256, 0, stream>>>(head, split, ws);

  dim3 grid(MAX_TILES, OUT_F / 128);
  lms_gemm<<<grid, 256, 0, stream>>>(input, weight, delta, bias, ws, out);
}